// HBV_function_74921409511427
// MI455X (gfx1250) — compile-verified
//
#include <hip/hip_runtime.h>

#define BLK 256

// ---------------------------------------------------------------------------
// HBV single-timestep flux kernel for MI455X (gfx1250).
//
// Each 256-thread block owns 256 batch elements. Inputs theta(256x13),
// y(256x5), climate[t](256x3) are staged into LDS with gfx1250 async
// global->LDS loads (b128 bulk + b32 tail, ASYNCcnt), each thread gathers its
// row from LDS (strides 13/5/3 are coprime with 64 banks -> conflict-free),
// runs the elementwise HBV math, writes dS rows back to LDS, and the block
// drains dS with coalesced async LDS->global b128 stores.
// ---------------------------------------------------------------------------

__device__ __forceinline__ unsigned lds_off(const void* p) {
    // Generic pointers into LDS carry the LDS byte offset in their low 32
    // bits.
    return (unsigned)(unsigned long long)p;
}

__device__ __forceinline__ void async_ld_b32(unsigned lds, const float* g) {
    asm volatile("global_load_async_to_lds_b32 %0, %1, off"
                 :: "v"(lds), "v"(g) : "memory");
}
__device__ __forceinline__ void async_ld_b128(unsigned lds, const float* g) {
    asm volatile("global_load_async_to_lds_b128 %0, %1, off"
                 :: "v"(lds), "v"(g) : "memory");
}
__device__ __forceinline__ void async_st_b32(float* g, unsigned lds) {
    asm volatile("global_store_async_from_lds_b32 %0, %1, off"
                 :: "v"(g), "v"(lds) : "memory");
}
__device__ __forceinline__ void async_st_b128(float* g, unsigned lds) {
    asm volatile("global_store_async_from_lds_b128 %0, %1, off"
                 :: "v"(g), "v"(lds) : "memory");
}
__device__ __forceinline__ void wait_async0() {
    asm volatile("s_wait_asynccnt 0" ::: "memory");
}

// Stage ndw dwords from g[] into contiguous LDS s[]; coalesced across lanes.
__device__ __forceinline__ void stage_in(float* s, const float* g, int ndw,
                                         bool aligned16) {
    const int tid = threadIdx.x;
    const int n4 = aligned16 ? (ndw >> 2) : 0;
    for (int i = tid; i < n4; i += BLK)
        async_ld_b128(lds_off(&s[i * 4]), g + (size_t)i * 4);
    for (int i = n4 * 4 + tid; i < ndw; i += BLK)
        async_ld_b32(lds_off(&s[i]), g + i);
}

__device__ __forceinline__ float pow_pos(float x, float p) {
    // x > 0 guaranteed; hardware transcendentals (v_log_f32 / v_exp_f32).
    return __builtin_amdgcn_exp2f(p * __builtin_amdgcn_logf(x));
}

__global__ __launch_bounds__(BLK) void hbv_step_kernel(
    const float* __restrict__ y,        // (B,5)
    const float* __restrict__ theta,    // (B,13)
    const float* __restrict__ clim,     // (T,B,3)
    const int*   __restrict__ tptr,     // scalar t
    float* __restrict__ out,            // dS (B,5) then fluxes (B,1)
    int B)
{
    __shared__ float s_th [BLK * 13];
    __shared__ float s_y  [BLK * 5];
    __shared__ float s_c  [BLK * 3];
    __shared__ float s_out[BLK * 5];

    const int tid = threadIdx.x;
    const int b0  = blockIdx.x * BLK;
    const int nv  = (B - b0 < BLK) ? (B - b0) : BLK;   // valid elems in block
    const int t   = tptr[0];

    const float* gth = theta + (size_t)b0 * 13;
    const float* gy  = y     + (size_t)b0 * 5;
    const float* gc  = clim  + ((size_t)t * (size_t)B + (size_t)b0) * 3;
    float*       go  = out   + (size_t)b0 * 5;

    const bool al_in  = ((((unsigned long long)gth | (unsigned long long)gy |
                           (unsigned long long)gc) & 15ull) == 0);
    const bool al_out = (((unsigned long long)go & 15ull) == 0);

    // ---- coalesced async staging: global -> LDS -------------------------
    stage_in(s_th, gth, nv * 13, al_in);
    stage_in(s_y,  gy,  nv * 5,  al_in);
    stage_in(s_c,  gc,  nv * 3,  al_in);

    wait_async0();        // this wave's async LDS writes landed
    __syncthreads();      // all waves' writes visible

    // ---- per-element HBV flux math --------------------------------------
    if (tid < nv) {
        const int b = b0 + tid;
        const float* th = &s_th[tid * 13];
        const float* sy = &s_y [tid * 5];
        const float* sc = &s_c [tid * 3];

        // Parameter unscaling: lo + theta*(hi-lo)
        const float Beta   = 1.0f   + th[0]  * 5.0f;
        const float FC     = 50.0f  + th[1]  * 950.0f;
        const float K0     = 0.05f  + th[2]  * 0.85f;
        const float K1     = 0.01f  + th[3]  * 0.49f;
        const float K2     = 0.001f + th[4]  * 0.199f;
        const float LP     = 0.2f   + th[5]  * 0.8f;
        const float PERC   =          th[6]  * 10.0f;
        const float UZL    =          th[7]  * 100.0f;
        const float TT     = -2.5f  + th[8]  * 5.0f;
        const float CFMAX  = 0.5f   + th[9]  * 9.5f;
        const float CFR    =          th[10] * 0.1f;
        const float CWH    =          th[11] * 0.2f;
        const float BETAET = 0.3f   + th[12] * 4.7f;

        const float SNOWPACK = fmaxf(sy[0], 0.0f);
        const float MELT     = fmaxf(sy[1], 0.0f);
        const float SM       = fmaxf(sy[2], 1e-8f);
        const float SUZ      = fmaxf(sy[3], 0.0f);
        const float SLZ      = fmaxf(sy[4], 0.0f);

        const float P  = sc[0];
        const float T_ = sc[1];
        const float Ep = sc[2];

        const float flux_sf    = (T_ < TT) ? P : 0.0f;
        const float flux_refr  = fminf(fmaxf(CFR * CFMAX * (TT - T_), 0.0f), MELT);
        const float flux_melt  = fminf(fmaxf(CFMAX * (T_ - TT), 0.0f), SNOWPACK);
        const float flux_rf    = (T_ >= TT) ? P : 0.0f;
        const float flux_Isnow = fmaxf(MELT - CWH * SNOWPACK, 0.0f);

        const float soil_wet   = fminf(fmaxf(pow_pos(SM / FC, Beta), 0.0f), 1.0f);
        const float flux_PEFF  = (flux_rf + flux_Isnow) * soil_wet;
        const float flux_ex    = fmaxf(SM - FC, 0.0f);
        const float evapf      = fminf(fmaxf(pow_pos(SM / (LP * FC), BETAET), 0.0f), 1.0f);
        const float flux_et    = fminf(SM, Ep * evapf);
        const float flux_perc  = fminf(SUZ, PERC);
        const float flux_q0    = K0 * fmaxf(SUZ - UZL, 0.0f);
        const float flux_q1    = K1 * SUZ;
        const float flux_q2    = K2 * SLZ;

        float* so = &s_out[tid * 5];                 // stride 5: conflict-free
        so[0] = flux_sf + flux_refr - flux_melt;
        so[1] = flux_melt - flux_refr - flux_Isnow;
        so[2] = flux_Isnow + flux_rf - flux_PEFF - flux_ex - flux_et;
        so[3] = flux_PEFF + flux_ex - flux_perc - flux_q0 - flux_q1;
        so[4] = flux_perc - flux_q2;

        // fluxes output: already perfectly coalesced per-lane b32 store
        out[(size_t)5 * (size_t)B + (size_t)b] = flux_q0 + flux_q1 + flux_q2;
    }

    __syncthreads();      // dS rows complete in LDS

    // ---- coalesced async drain: LDS -> global (dS) ----------------------
    {
        const int ndw = nv * 5;
        const int n4  = al_out ? (ndw >> 2) : 0;
        for (int i = tid; i < n4; i += BLK)
            async_st_b128(go + (size_t)i * 4, lds_off(&s_out[i * 4]));
        for (int i = n4 * 4 + tid; i < ndw; i += BLK)
            async_st_b32(go + i, lds_off(&s_out[i]));
    }
    // S_ENDPGM performs an implicit wait-idle, draining ASYNCcnt.
}

extern "C" void kernel_launch(void* const* d_in, const int* in_sizes, int n_in,
                              void* d_out, int out_size, void* d_ws, size_t ws_size,
                              hipStream_t stream) {
    const float* y     = (const float*)d_in[0];
    const float* theta = (const float*)d_in[1];
    const float* clim  = (const float*)d_in[2];
    const int*   tptr  = (const int*)  d_in[3];
    float*       out   = (float*)d_out;

    const int B = in_sizes[0] / 5;   // y is (B,5)
    const int grid = (B + BLK - 1) / BLK;

    hbv_step_kernel<<<grid, BLK, 0, stream>>>(y, theta, clim, tptr, out, B);
}